// ForwardRadonLayer_36696200577567
// MI455X (gfx1250) — compile-verified
//
#include <hip/hip_runtime.h>

#define N_ANGLES   180
#define N_DET      256
#define IMG_H      256
#define IMG_W      256
#define N_T        256
#define LDS_STRIDE 260   // 260 % 64 == 4 -> vertical neighbors hit distinct LDS banks

typedef __attribute__((ext_vector_type(2))) float    v2f;
typedef __attribute__((ext_vector_type(8))) float    v8f;
typedef __attribute__((ext_vector_type(4))) unsigned v4u;
typedef __attribute__((ext_vector_type(8))) int      v8i;
typedef __attribute__((ext_vector_type(4))) int      v4i;

// Branchless bilinear sample from LDS-staged image; zero outside [0,256)^2.
// Loads are unconditional from clamped (in-range) addresses; validity is
// folded into the interpolation weights -> v_cndmask, no exec-mask blocks.
__device__ __forceinline__ float bsample(const float* lds, float x, float y) {
    float xf = floorf(x), yf = floorf(y);
    float wx = x - xf,    wy = y - yf;
    int ix = (int)xf,     iy = (int)yf;
    int ix0 = min(max(ix, 0), IMG_W - 1);
    int ix1 = min(max(ix + 1, 0), IMG_W - 1);
    int iy0 = min(max(iy, 0), IMG_H - 1);
    int iy1 = min(max(iy + 1, 0), IMG_H - 1);
    float fx0 = ((unsigned)ix       < (unsigned)IMG_W) ? (1.0f - wx) : 0.0f;
    float fx1 = ((unsigned)(ix + 1) < (unsigned)IMG_W) ? wx          : 0.0f;
    float fy0 = ((unsigned)iy       < (unsigned)IMG_H) ? (1.0f - wy) : 0.0f;
    float fy1 = ((unsigned)(iy + 1) < (unsigned)IMG_H) ? wy          : 0.0f;
    const float* r0 = lds + iy0 * LDS_STRIDE;
    const float* r1 = lds + iy1 * LDS_STRIDE;
    float v00 = r0[ix0], v01 = r0[ix1];
    float v10 = r1[ix0], v11 = r1[ix1];
    float top = v00 * fx0 + v01 * fx1;
    float bot = v10 * fx0 + v11 * fx1;
    return top * fy0 + bot * fy1;
}

__global__ __launch_bounds__(256)
void radon_fwd_kernel(const float* __restrict__ img, float* __restrict__ out) {
    __shared__ float lds[IMG_H * LDS_STRIDE];   // 260 KB of the 320 KB WGP LDS

    const int blk = blockIdx.x;                 // blk = b * N_ANGLES + a
    const int b   = blk / N_ANGLES;
    const int tid = threadIdx.x;
    const int a   = blk - b * N_ANGLES;

    // ---- Stage image b into LDS with one Tensor Data Mover descriptor. ----
    // TDM pad_enable inserts 4 DWORDs of padding after every 256 DWORDs,
    // producing the 260-float padded row stride in hardware.
    {
        const unsigned lds_base = (unsigned)(size_t)(void*)lds;  // shared aperture: low 32 bits = LDS byte offset
        const unsigned long long ga =
            (unsigned long long)(size_t)(img + (size_t)b * (IMG_H * IMG_W));

        v4u g0;
        g0.x = 1u;                                   // count=1 (valid), is_restore=0, gather off
        g0.y = lds_base;                             // lds_addr (bytes)
        g0.z = (unsigned)ga;                         // global_addr[31:0]   -> D#[95:64]
        g0.w = ((unsigned)(ga >> 32) & 0x01FFFFFFu)  // global_addr[56:32]  -> D#[120:96]
             | (2u << 30);                           // type=2 ("image")    -> D#[127:126]

        v8i g1;
        g1[0] = (int)((2u << 16)      // data_size = 4 bytes
                    | (1u << 20)      // pad_enable
                    | (5u << 22)      // pad_interval: 256 DWORDs (one 256-float row)
                    | (3u << 25));    // pad_amount:   4 DWORDs  (row stride -> 260)
        g1[1] = (int)(256u << 16);    // tensor_dim0[15:0] = 256  (bits 63:48)
        g1[2] = (int)(256u << 16);    // tensor_dim1[15:0] = 256  (bits 95:80)
        g1[3] = (int)(256u << 16);    // tile_dim0 = 256          (bits 127:112)
        g1[4] = (int)256u;            // tile_dim1 = 256, tile_dim2 = 0 (unused)
        g1[5] = (int)256u;            // tensor_dim0_stride = 256
        g1[6] = 0;                    // stride hi / tensor_dim1_stride (unused, 2D)
        g1[7] = 0;
        v4i g2 = {0, 0, 0, 0};        // dims 3/4 unused
        v4i g3 = {0, 0, 0, 0};
        v8i gx = {0, 0, 0, 0, 0, 0, 0, 0};  // clang-23 6-arg form: extra group, zero-filled

        if (tid < 32) {               // one wave issues the DMA; EXEC is ignored by TDM
            __builtin_amdgcn_tensor_load_to_lds(g0, g1, g2, g3, gx, 0);
            __builtin_amdgcn_s_wait_tensorcnt(0);
        }
        asm volatile("" ::: "memory");
    }
    __syncthreads();

    // ---- Ray geometry ----
    const float theta = (float)a * 0.017453292519943295f;   // pi/180
    float sn, cs;
    __sincosf(theta, &sn, &cs);

    const int lane  = tid & 31;
    const int wave  = tid >> 5;
    const int m     = lane & 15;     // WMMA row (detector within the 16-ray tile)
    const int khalf = lane >> 4;     // lanes 0-15: K=0,1 ; lanes 16-31: K=2,3

    v2f bones; bones.x = 1.0f; bones.y = 1.0f;  // all-ones B => D[m,n] = sum_k A[m,k] + C

    const float dx1 = -sn,        dy1 = cs;
    const float dx4 = -4.0f * sn, dy4 = 4.0f * cs;

    #pragma unroll
    for (int pass = 0; pass < 2; ++pass) {
        const int   dbase = pass * 128 + wave * 16;
        const float s     = (float)(dbase + m) - 127.5f;

        // Chain 0 walks t in [0,128), chain 1 walks t in [128,256):
        // two independent WMMA accumulator chains to break the D->C dependency.
        const float t0 = -127.5f + (float)(khalf << 1);
        float xa = 127.5f + s * cs - t0 * sn;
        float ya = 127.5f + s * sn + t0 * cs;
        float xb = xa + 128.0f * dx1;
        float yb = ya + 128.0f * dy1;

        v8f c0 = {0.f, 0.f, 0.f, 0.f, 0.f, 0.f, 0.f, 0.f};
        v8f c1 = {0.f, 0.f, 0.f, 0.f, 0.f, 0.f, 0.f, 0.f};

        #pragma unroll 2
        for (int it = 0; it < N_T / 8; ++it) {   // 32 iterations, 8 t-steps each
            v2f a0;
            a0.x = bsample(lds, xa,       ya);
            a0.y = bsample(lds, xa + dx1, ya + dy1);
            c0 = __builtin_amdgcn_wmma_f32_16x16x4_f32(
                     false, a0, false, bones, (short)0, c0, false, false);
            v2f a1;
            a1.x = bsample(lds, xb,       yb);
            a1.y = bsample(lds, xb + dx1, yb + dy1);
            c1 = __builtin_amdgcn_wmma_f32_16x16x4_f32(
                     false, a1, false, bones, (short)0, c1, false, false);
            xa += dx4; ya += dy4;
            xb += dx4; yb += dy4;
        }

        v8f c = c0 + c1;

        // Column N=0 of D: lane 0 holds rows M=0..7, lane 16 holds rows M=8..15.
        if (m == 0) {
            float* o = out + (size_t)blk * N_DET + dbase + khalf * 8;
            #pragma unroll
            for (int r = 0; r < 8; ++r) o[r] = c[r];
        }
    }
}

extern "C" void kernel_launch(void* const* d_in, const int* in_sizes, int n_in,
                              void* d_out, int out_size, void* d_ws, size_t ws_size,
                              hipStream_t stream) {
    const float* img = (const float*)d_in[0];
    float* out = (float*)d_out;
    const int B = in_sizes[0] / (IMG_H * IMG_W);   // = 8
    dim3 grid(B * N_ANGLES);                        // one (image, angle) pair per workgroup
    dim3 block(256);                                // 8 wave32s
    radon_fwd_kernel<<<grid, block, 0, stream>>>(img, out);
}